// FSQ_Transformer_14834817040786
// MI455X (gfx1250) — compile-verified
//
#include <hip/hip_runtime.h>
#include <hip/hip_bf16.h>
#include <math.h>

// ---------------- problem constants (from the reference) ----------------
#define BATCH        16
#define SEQ          2048
#define DMODEL       1024
#define ELAT         5
#define NTOK         (BATCH * SEQ)          // 32768
#define LATENT_SEQ   1024
#define KCODE        4375                   // 7*5*5*5*5

// d_out float offsets
#define OFF_XOUT     0
#define SZ_XOUT      (NTOK * DMODEL)        // 33554432
#define OFF_ZE       (OFF_XOUT + SZ_XOUT)
#define SZ_Z         (NTOK * ELAT)          // 163840
#define OFF_ZQ       (OFF_ZE + SZ_Z)
#define OFF_USAGE    (OFF_ZQ + SZ_Z)
#define OFF_UNIQUE   (OFF_USAGE + 1)
#define OFF_PM       (OFF_UNIQUE + 1)

typedef __attribute__((ext_vector_type(2))) float v2f;
typedef __attribute__((ext_vector_type(8))) float v8f;

// ---------------------------------------------------------------------------
// Kernel 0: zero the codebook-usage bitmap in workspace (harness won't).
// ---------------------------------------------------------------------------
__global__ void fsq_zero_used(int* __restrict__ used) {
    int i = blockIdx.x * blockDim.x + threadIdx.x;
    if (i < KCODE) used[i] = 0;
}

// ---------------------------------------------------------------------------
// Kernel 1: encode + mask + FSQ quantize + codebook index mark.
// One wave32 per token: lanes stripe the D=1024 contraction (coalesced 128B
// per wave per iteration), W_enc (20KB) staged in LDS (bank-conflict-free:
// word index 5j+e mod 64 distinct for 32 lanes since gcd(5,64)=1), then a
// shfl_xor tree reduce. 8 waves / 256-thread block, 4096 blocks.
// ---------------------------------------------------------------------------
__global__ __launch_bounds__(256) void fsq_encode(
    const float* __restrict__ x,      // [NTOK, DMODEL]
    const float* __restrict__ Wenc,   // [DMODEL, ELAT]
    float* __restrict__ zE,           // [NTOK, ELAT]
    float* __restrict__ zQ,           // [NTOK, ELAT]
    int*   __restrict__ used)         // [KCODE]
{
    __shared__ float sW[DMODEL * ELAT]; // 20 KB of the 320 KB WGP LDS
    for (int i = threadIdx.x; i < DMODEL * ELAT; i += 256) sW[i] = Wenc[i];
    __syncthreads();

    const int wave = threadIdx.x >> 5;
    const int lane = threadIdx.x & 31;
    const int t    = blockIdx.x * 8 + wave;            // token id, exact cover

    const float* __restrict__ xr = x + (size_t)t * DMODEL;
    float a0 = 0.f, a1 = 0.f, a2 = 0.f, a3 = 0.f, a4 = 0.f;
    #pragma unroll 4
    for (int j = lane; j < DMODEL; j += 32) {
        const float xv = xr[j];
        const float* w = &sW[j * ELAT];
        a0 = fmaf(xv, w[0], a0);
        a1 = fmaf(xv, w[1], a1);
        a2 = fmaf(xv, w[2], a2);
        a3 = fmaf(xv, w[3], a3);
        a4 = fmaf(xv, w[4], a4);
    }
    #pragma unroll
    for (int off = 16; off > 0; off >>= 1) {           // wave32 tree reduce
        a0 += __shfl_xor(a0, off, 32);
        a1 += __shfl_xor(a1, off, 32);
        a2 += __shfl_xor(a2, off, 32);
        a3 += __shfl_xor(a3, off, 32);
        a4 += __shfl_xor(a4, off, 32);
    }

    if (lane == 0) {
        const int   s = t & (SEQ - 1);
        const float m = (s < LATENT_SEQ) ? 1.0f : 0.0f;
        float h[ELAT] = { a0 * m, a1 * m, a2 * m, a3 * m, a4 * m };
        const float halfL[ELAT]  = { 3.f, 2.f, 2.f, 2.f, 2.f };
        const int   halfI[ELAT]  = { 3, 2, 2, 2, 2 };
        const int   strideI[ELAT] = { 625, 125, 25, 5, 1 }; // meshgrid 'ij'
        int idx = 0;
        #pragma unroll
        for (int e = 0; e < ELAT; ++e) {
            zE[(size_t)t * ELAT + e] = h[e];           // z_e = masked encoder out
            const float zsq = halfL[e] * tanhf(h[e]);
            const float zq  = rintf(zsq);              // round half-to-even
            zQ[(size_t)t * ELAT + e] = zq;             // forward value of STE z
            idx += ((int)zq + halfI[e]) * strideI[e];
        }
        used[idx] = 1;                                 // racing 1-stores: benign
    }
}

// ---------------------------------------------------------------------------
// Kernel 2: reduce the usage bitmap -> unique, usage, percent_masked.
// ---------------------------------------------------------------------------
__global__ __launch_bounds__(256) void fsq_usage(
    const int* __restrict__ used,
    float* __restrict__ out_usage,
    float* __restrict__ out_unique,
    float* __restrict__ out_pm)
{
    __shared__ int red[256];
    int s = 0;
    for (int i = threadIdx.x; i < KCODE; i += 256) s += used[i];
    red[threadIdx.x] = s;
    __syncthreads();
    for (int step = 128; step > 0; step >>= 1) {
        if (threadIdx.x < step) red[threadIdx.x] += red[threadIdx.x + step];
        __syncthreads();
    }
    if (threadIdx.x == 0) {
        const float uniq = (float)red[0];
        const float K    = (float)KCODE;
        *out_usage  = (uniq + (K - uniq) * expf(-1.0f)) / K;
        *out_unique = uniq;                            // int output stored as f32
        *out_pm     = 0.0f;
    }
}

// ---------------------------------------------------------------------------
// Kernel 3: decode x_out = z @ W_dec with V_WMMA_F32_16X16X4_F32.
// z is [NTOK,5]; K=5 padded to 8 -> two wmma ops per 16x16 output tile.
// One block per 16-token tile (2048 blocks); each of the 8 waves owns 128
// output columns (8 col-tiles) -> 16 wmma per wave, 262144 wmma total.
// A fragment (16x4 f32): lane%16 = M (token row), lane/16 selects K pair,
//   vgpr0/1 = K{0,1} (lanes 0-15) / K{2,3} (lanes 16-31).
// B fragment (4x16 f32): lane%16 = N (column), same K striping.
// C/D fragment (8 VGPRs): vgpr r -> row r (lanes 0-15) / row r+8 (16-31).
// Pad lanes use UNCONDITIONAL loads + lane-select (v_cndmask), never
// conditional loads, so no EXEC save/restore branches appear in the hot loop
// and EXEC stays all-ones as WMMA requires.
// ---------------------------------------------------------------------------
__global__ __launch_bounds__(256) void fsq_decode(
    const float* __restrict__ z,      // [NTOK, ELAT] (z_q region of d_out)
    const float* __restrict__ Wdec,   // [ELAT, DMODEL]
    float* __restrict__ xout)         // [NTOK, DMODEL]
{
    const int wave = threadIdx.x >> 5;
    const int lane = threadIdx.x & 31;
    const int l16  = lane & 15;
    const int hi   = lane >> 4;                  // 0: lanes 0-15, 1: lanes 16-31
    const float loMask = (hi == 0) ? 1.0f : 0.0f;

    const int tokBase = blockIdx.x * 16;
    const int tok     = tokBase + l16;

    // ---- A fragments: z tile, K = 0..3 and K = 4..7 (5..7 zero-padded) ----
    const float* zr = z + (size_t)tok * ELAT;
    const int kb = hi * 2;                       // 0 or 2
    v2f aLo, aHi;
    aLo[0] = zr[kb];
    aLo[1] = zr[kb + 1];
    aHi[0] = zr[4] * loMask;                     // K=4 real, K=6 pad (select, no branch)
    aHi[1] = 0.0f;                               // K=5 / K=7 pad

    const float* __restrict__ WdRow4 = Wdec + (size_t)4 * DMODEL;
    const int colBase0 = wave * 128;
    #pragma unroll
    for (int ct = 0; ct < 8; ++ct) {
        const int col = colBase0 + ct * 16 + l16;

        // ---- B fragments: W_dec rows K (4x16 tile), K striped like A ----
        // Row-4 element loaded by ALL lanes (valid address), zeroed by select
        // in the upper half -> pure v_cndmask, no divergent branch.
        v2f bLo, bHi;
        bLo[0] = Wdec[(size_t)kb * DMODEL + col];
        bLo[1] = Wdec[(size_t)(kb + 1) * DMODEL + col];
        bHi[0] = WdRow4[col] * loMask;
        bHi[1] = 0.0f;

        v8f c = {};
        c = __builtin_amdgcn_wmma_f32_16x16x4_f32(
                false, aLo, false, bLo, (short)0, c, false, false);
        c = __builtin_amdgcn_wmma_f32_16x16x4_f32(
                false, aHi, false, bHi, (short)0, c, false, false);

        // ---- store C: vgpr r -> token row r (lo half) / r+8 (hi half) ----
        #pragma unroll
        for (int r = 0; r < 8; ++r) {
            const int row = r + hi * 8;
            xout[(size_t)(tokBase + row) * DMODEL + col] = c[r];
        }
    }
}

// ---------------------------------------------------------------------------
extern "C" void kernel_launch(void* const* d_in, const int* in_sizes, int n_in,
                              void* d_out, int out_size, void* d_ws, size_t ws_size,
                              hipStream_t stream) {
    (void)in_sizes; (void)n_in; (void)out_size; (void)ws_size;

    const float* x    = (const float*)d_in[0];   // [16,2048,1024]
    const float* Wenc = (const float*)d_in[1];   // [1024,5]
    const float* Wdec = (const float*)d_in[2];   // [5,1024]

    float* out    = (float*)d_out;
    float* xout   = out + OFF_XOUT;
    float* zE     = out + OFF_ZE;
    float* zQ     = out + OFF_ZQ;
    float* usage  = out + OFF_USAGE;
    float* unique = out + OFF_UNIQUE;
    float* pm     = out + OFF_PM;

    int* used = (int*)d_ws;                      // KCODE ints

    fsq_zero_used<<<(KCODE + 255) / 256, 256, 0, stream>>>(used);
    fsq_encode<<<NTOK / 8, 256, 0, stream>>>(x, Wenc, zE, zQ, used);
    fsq_usage<<<1, 256, 0, stream>>>(used, usage, unique, pm);
    fsq_decode<<<NTOK / 16, 256, 0, stream>>>(zQ, Wdec, xout);
}